// GATV2_44693429682811
// MI455X (gfx1250) — compile-verified
//
#include <hip/hip_runtime.h>
#include <math.h>

// ---------------------------------------------------------------------------
// GATv2 (heads=1), 3 layers, N=100000 nodes, E=1600000 edges, 64 channels.
// Dense GEMMs -> V_WMMA_F32_16X16X4_F32 (exact fp32 semantics).
// Edge phase  -> L2-resident gathers + f32 atomics (all node tensors < 192MB L2).
// ---------------------------------------------------------------------------

#define N_NODES   100000
#define N_EDGES   1600000
#define CH        64
#define NEG_SLOPE 0.2f
#define NE_TOT    (N_EDGES + N_NODES)   // edges + self loops

typedef float v2f __attribute__((ext_vector_type(2)));
typedef float v8f __attribute__((ext_vector_type(8)));

// ---- float atomic max via monotone int mapping (lowers to L2 int atomics) --
__device__ __forceinline__ void atomic_max_f32(float* addr, float val) {
    if (val >= 0.0f) atomicMax((int*)addr, __float_as_int(val));
    else             atomicMin((unsigned int*)addr, __float_as_uint(val));
}

// ---------------------------------------------------------------------------
// Fused GEMM: XL = X@Wl + bl ; XR = X@Wr + br.   X: N x 64, W: 64 x 64.
// Block = 256 threads = 8 waves; each wave owns a 16-row tile and produces
// both 16x64 outputs via 2 (mats) x 4 (col tiles) x 16 (K steps) WMMAs.
// Weights staged in LDS; A-fragments (the wave's 16x64 row slab) cached in
// VGPRs and reused for both weight matrices.
// ---------------------------------------------------------------------------
__global__ __launch_bounds__(256)
void gemm_xl_xr_kernel(const float* __restrict__ X,
                       const float* __restrict__ Wl, const float* __restrict__ bl,
                       const float* __restrict__ Wr, const float* __restrict__ br,
                       float* __restrict__ XL, float* __restrict__ XR)
{
    __shared__ float sW[2 * CH * CH];                 // 32 KB: [Wl | Wr]
    {
        const int t = threadIdx.x;
        const float4* gl = (const float4*)Wl;
        const float4* gr = (const float4*)Wr;
        float4* s0 = (float4*)&sW[0];
        float4* s1 = (float4*)&sW[CH * CH];
        #pragma unroll
        for (int i = t; i < (CH * CH) / 4; i += 256) { s0[i] = gl[i]; s1[i] = gr[i]; }
    }
    __syncthreads();

    const int wave = threadIdx.x >> 5;
    const int lane = threadIdx.x & 31;
    const int m    = lane & 15;        // row-within-tile (A) / col-within-tile (B,C)
    const int kh   = lane >> 4;        // half-wave selector
    const int row0 = (blockIdx.x * 8 + wave) * 16;
    if (row0 >= N_NODES) return;       // wave-uniform: EXEC stays all-ones for WMMA

    // A fragments for the whole 16x64 row slab: 16 K-steps, v2f each.
    // Layout (ISA 7.12.2, 32-bit A 16x4): lane(kh*16+m) regs = A[m][k0+2kh], A[m][k0+2kh+1]
    v2f afrag[16];
    const float* xrow = X + (size_t)(row0 + m) * CH;
    #pragma unroll
    for (int ks = 0; ks < 16; ++ks) {
        const int k = ks * 4 + 2 * kh;
        afrag[ks][0] = xrow[k];
        afrag[ks][1] = xrow[k + 1];
    }

    #pragma unroll
    for (int sel = 0; sel < 2; ++sel) {
        const float* sw   = &sW[sel * CH * CH];
        const float* bias = sel ? br : bl;
        float*       outp = sel ? XR : XL;
        #pragma unroll
        for (int nt = 0; nt < 4; ++nt) {
            const int col = nt * 16 + m;
            v8f acc = {};
            #pragma unroll
            for (int ks = 0; ks < 16; ++ks) {
                const int krow = ks * 4 + 2 * kh;   // B: lane holds B[krow][col], B[krow+1][col]
                v2f b;
                b[0] = sw[krow * CH + col];
                b[1] = sw[(krow + 1) * CH + col];
                acc = __builtin_amdgcn_wmma_f32_16x16x4_f32(
                          false, afrag[ks], false, b, (short)0, acc, false, false);
            }
            const float bv = bias[col];
            #pragma unroll
            for (int v = 0; v < 8; ++v) {           // C/D: row = v + 8*kh, col = lane&15
                outp[(size_t)(row0 + v + 8 * kh) * CH + nt * 16 + m] = acc[v] + bv;
            }
        }
    }
}

// ---------------------------------------------------------------------------
// Self-loop attr = mean incoming edge weight (fill_value='mean').
// ---------------------------------------------------------------------------
__global__ void zero2_kernel(float* __restrict__ a, float* __restrict__ b) {
    int i = blockIdx.x * blockDim.x + threadIdx.x;
    if (i < N_NODES) { a[i] = 0.0f; b[i] = 0.0f; }
}

__global__ void edge_degree_kernel(const int* __restrict__ ei, const float* __restrict__ ew,
                                   float* __restrict__ cnt, float* __restrict__ sw) {
    int e = blockIdx.x * blockDim.x + threadIdx.x;
    if (e >= N_EDGES) return;
    int dst = ei[N_EDGES + e];
    atomicAdd(&cnt[dst], 1.0f);
    atomicAdd(&sw[dst],  ew[e]);
}

__global__ void loop_attr_kernel(float* __restrict__ sw, const float* __restrict__ cnt) {
    int i = blockIdx.x * blockDim.x + threadIdx.x;
    if (i < N_NODES) sw[i] = sw[i] / fmaxf(cnt[i], 1.0f);   // in place -> loop_attr
}

// ---------------------------------------------------------------------------
// Per-layer reset: acc = 0, mx = -inf, den = 0.
// ---------------------------------------------------------------------------
__global__ void layer_init_kernel(float* __restrict__ acc, float* __restrict__ mx,
                                  float* __restrict__ den) {
    int i = blockIdx.x * blockDim.x + threadIdx.x;
    if (i < N_NODES * CH) acc[i] = 0.0f;
    if (i < N_NODES) { mx[i] = -INFINITY; den[i] = 0.0f; }
}

// ---------------------------------------------------------------------------
// Pass 1: logit_e = att . leaky_relu(xl[src] + xr[dst] + ea*We); segment max.
// One thread per edge; 16 float4 gathers per side (L2-resident).
// ---------------------------------------------------------------------------
__global__ __launch_bounds__(256)
void edge_logit_kernel(const int* __restrict__ ei, const float* __restrict__ ew,
                       const float* __restrict__ loop_attr,
                       const float* __restrict__ XL, const float* __restrict__ XR,
                       const float* __restrict__ We, const float* __restrict__ att,
                       float* __restrict__ lbuf, float* __restrict__ mx, int has_ea)
{
    int e = blockIdx.x * blockDim.x + threadIdx.x;
    if (e >= NE_TOT) return;
    int src, dst; float ea;
    if (e < N_EDGES) { src = ei[e]; dst = ei[N_EDGES + e]; ea = has_ea ? ew[e] : 0.0f; }
    else             { src = dst = e - N_EDGES;            ea = has_ea ? loop_attr[src] : 0.0f; }

    const float4* xl4 = (const float4*)(XL + (size_t)src * CH);
    const float4* xr4 = (const float4*)(XR + (size_t)dst * CH);
    const float4* we4 = (const float4*)We;
    const float4* at4 = (const float4*)att;

    float logit = 0.0f;
    #pragma unroll
    for (int q = 0; q < CH / 4; ++q) {
        float4 a = xl4[q], b = xr4[q], w = we4[q], t = at4[q];
        float m0 = a.x + b.x + ea * w.x;
        float m1 = a.y + b.y + ea * w.y;
        float m2 = a.z + b.z + ea * w.z;
        float m3 = a.w + b.w + ea * w.w;
        m0 = (m0 > 0.0f) ? m0 : NEG_SLOPE * m0;
        m1 = (m1 > 0.0f) ? m1 : NEG_SLOPE * m1;
        m2 = (m2 > 0.0f) ? m2 : NEG_SLOPE * m2;
        m3 = (m3 > 0.0f) ? m3 : NEG_SLOPE * m3;
        logit += m0 * t.x + m1 * t.y + m2 * t.z + m3 * t.w;
    }
    lbuf[e] = logit;
    atomic_max_f32(&mx[dst], logit);
}

// ---------------------------------------------------------------------------
// Pass 2: ex = exp(logit - mx[dst]); den[dst] += ex.  (ex overwrites lbuf)
// ---------------------------------------------------------------------------
__global__ void edge_exp_kernel(const int* __restrict__ ei, float* __restrict__ lbuf,
                                const float* __restrict__ mx, float* __restrict__ den)
{
    int e = blockIdx.x * blockDim.x + threadIdx.x;
    if (e >= NE_TOT) return;
    int dst = (e < N_EDGES) ? ei[N_EDGES + e] : (e - N_EDGES);
    float ex = __expf(lbuf[e] - mx[dst]);
    lbuf[e] = ex;
    atomicAdd(&den[dst], ex);
}

// ---------------------------------------------------------------------------
// Pass 3: acc[dst] += (ex/den[dst]) * xl[src].  32 lanes cover one edge's
// 64 channels (float2 per lane) -> coalesced gather + paired f32 atomics.
// ---------------------------------------------------------------------------
__global__ __launch_bounds__(256)
void edge_accum_kernel(const int* __restrict__ ei, const float* __restrict__ lbuf,
                       const float* __restrict__ den, const float* __restrict__ XL,
                       float* __restrict__ acc)
{
    long long tid = (long long)blockIdx.x * blockDim.x + threadIdx.x;
    int e = (int)(tid >> 5);
    if (e >= NE_TOT) return;
    int c = ((int)tid & 31) * 2;
    int src, dst;
    if (e < N_EDGES) { src = ei[e]; dst = ei[N_EDGES + e]; }
    else             { src = dst = e - N_EDGES; }
    float alpha = lbuf[e] / den[dst];
    float2 v = *(const float2*)(XL + (size_t)src * CH + c);
    atomicAdd(&acc[(size_t)dst * CH + c],     alpha * v.x);
    atomicAdd(&acc[(size_t)dst * CH + c + 1], alpha * v.y);
}

// ---------------------------------------------------------------------------
// Finalize: out = acc + bias, optional relu.
// ---------------------------------------------------------------------------
__global__ void finalize_kernel(const float* __restrict__ acc, const float* __restrict__ bias,
                                float* __restrict__ out, int do_relu)
{
    int i = blockIdx.x * blockDim.x + threadIdx.x;
    if (i >= N_NODES * CH) return;
    float v = acc[i] + bias[i & (CH - 1)];
    if (do_relu) v = fmaxf(v, 0.0f);
    out[i] = v;
}

// ---------------------------------------------------------------------------
extern "C" void kernel_launch(void* const* d_in, const int* in_sizes, int n_in,
                              void* d_out, int out_size, void* d_ws, size_t ws_size,
                              hipStream_t stream)
{
    (void)in_sizes; (void)n_in; (void)out_size; (void)ws_size;

    const float* x  = (const float*)d_in[0];
    const int*   ei = (const int*)d_in[1];
    const float* ew = (const float*)d_in[2];
    // params flattened per layer in dict insertion order: Wl, bl, Wr, br, We, att, bias
    auto P = [&](int layer, int k) { return (const float*)d_in[3 + layer * 7 + k]; };

    char* ws = (char*)d_ws;
    size_t off = 0;
    auto carve = [&](size_t bytes) {
        void* p = ws + off;
        off += (bytes + 255) & ~(size_t)255;
        return p;
    };
    float* XL    = (float*)carve(sizeof(float) * (size_t)N_NODES * CH);
    float* XR    = (float*)carve(sizeof(float) * (size_t)N_NODES * CH);
    float* ACC   = (float*)carve(sizeof(float) * (size_t)N_NODES * CH);
    float* H     = (float*)carve(sizeof(float) * (size_t)N_NODES * CH);
    float* MX    = (float*)carve(sizeof(float) * N_NODES);
    float* DEN   = (float*)carve(sizeof(float) * N_NODES);
    float* CNT   = (float*)carve(sizeof(float) * N_NODES);
    float* LATTR = (float*)carve(sizeof(float) * N_NODES);   // sum-w, then mean
    float* LBUF  = (float*)carve(sizeof(float) * NE_TOT);

    const int B = 256;
    const int gN    = (N_NODES + B - 1) / B;            // 391
    const int gE    = (N_EDGES + B - 1) / B;            // 6250
    const int gNE   = (NE_TOT + B - 1) / B;             // 6641
    const int gNC   = (N_NODES * CH + B - 1) / B;       // 25000
    const int gACC  = (int)(((long long)NE_TOT * 32 + B - 1) / B);  // 212500
    const int gGEMM = ((N_NODES + 127) / 128);          // 8 row-tiles of 16 per block -> 782

    // Self-loop attr (same edge weights feed layers 0 and 1; layer 2 has none).
    zero2_kernel<<<gN, B, 0, stream>>>(CNT, LATTR);
    edge_degree_kernel<<<gE, B, 0, stream>>>(ei, ew, CNT, LATTR);
    loop_attr_kernel<<<gN, B, 0, stream>>>(LATTR, CNT);

    const float* hin = x;
    for (int layer = 0; layer < 3; ++layer) {
        const int has_ea = (layer < 2) ? 1 : 0;
        layer_init_kernel<<<gNC, B, 0, stream>>>(ACC, MX, DEN);
        gemm_xl_xr_kernel<<<gGEMM, B, 0, stream>>>(hin, P(layer, 0), P(layer, 1),
                                                   P(layer, 2), P(layer, 3), XL, XR);
        edge_logit_kernel<<<gNE, B, 0, stream>>>(ei, ew, LATTR, XL, XR,
                                                 P(layer, 4), P(layer, 5), LBUF, MX, has_ea);
        edge_exp_kernel<<<gNE, B, 0, stream>>>(ei, LBUF, MX, DEN);
        edge_accum_kernel<<<gACC, B, 0, stream>>>(ei, LBUF, DEN, XL, ACC);
        float* outp = (layer == 2) ? (float*)d_out : H;
        finalize_kernel<<<gNC, B, 0, stream>>>(ACC, P(layer, 6), outp, has_ea);
        hin = H;
    }
}